// SparseGraphNetwork_22797686408053
// MI455X (gfx1250) — compile-verified
//
#include <hip/hip_runtime.h>
#include <hip/hip_bf16.h>

typedef float v2f __attribute__((ext_vector_type(2)));
typedef float v8f __attribute__((ext_vector_type(8)));

#define N_NODES 50000
#define N_EDGES 800000
#define DIM     128
#define HEADD   64

// ---- monotone float <-> uint encoding for atomic segment-max ----
__device__ __forceinline__ unsigned encf(float f) {
    unsigned u = __float_as_uint(f);
    return (u & 0x80000000u) ? ~u : (u | 0x80000000u);
}
__device__ __forceinline__ float decf(unsigned e) {
    unsigned u = (e & 0x80000000u) ? (e & 0x7FFFFFFFu) : ~e;
    return __uint_as_float(u);
}
#define ENC_NEG_INF 0x007FFFFFu   // encf(-inf)

// ================= fp32 WMMA GEMM: C[N x COLS] = A[N x 128] @ W[128 x COLS] (+bias) ==========
// grid.x = N/16 ; block = (32, COLS/16). Each wave owns one 16-col tile.
template <int COLS>
__global__ __launch_bounds__(COLS * 2) void gemm_wmma(
    const float* __restrict__ A, const float* __restrict__ W,
    const float* __restrict__ bias, float* __restrict__ C)
{
    __shared__ __align__(16) float As[16 * 128];

    const int lane    = threadIdx.x;          // 0..31
    const int wave    = threadIdx.y;          // 0..COLS/16-1
    const int rowBase = blockIdx.x * 16;
    const int tid     = lane + 32 * wave;
    const int nthr    = 32 * (COLS / 16);

    // A-tile for rows [rowBase, rowBase+16) is a contiguous 2048-float block
    const float* Ablk = A + (size_t)rowBase * 128;
    for (int i = tid; i < 16 * 128; i += nthr) As[i] = Ablk[i];
    __syncthreads();

    const int col  = wave * 16 + (lane & 15);
    const int koff = (lane >> 4) << 1;   // lanes 0-15 -> K{0,1}, lanes 16-31 -> K{2,3}
    const int row  = lane & 15;

    v8f acc = {};
#pragma unroll
    for (int k = 0; k < 128; k += 4) {
        v2f a = *(const v2f*)(&As[row * 128 + k + koff]);
        v2f b;
        b.x = W[(k + koff) * COLS + col];
        b.y = W[(k + koff + 1) * COLS + col];
        acc = __builtin_amdgcn_wmma_f32_16x16x4_f32(
            /*neg_a=*/false, a, /*neg_b=*/false, b,
            /*c_mod=*/(short)0, acc, /*reuse_a=*/false, /*reuse_b=*/false);
    }

    const float bi   = bias ? bias[col] : 0.0f;
    const int   rOff = (lane >> 4) * 8;  // lanes>=16 hold rows M+8
#pragma unroll
    for (int i = 0; i < 8; ++i)
        C[(size_t)(rowBase + rOff + i) * COLS + col] = acc[i] + bi;
}

// ================= per-node dot: out[n] = H[n,:] . a ==================
__global__ __launch_bounds__(256) void dot_kernel(
    const float* __restrict__ H, const float* __restrict__ a, float* __restrict__ out)
{
    const int node = blockIdx.x * 8 + threadIdx.y;
    const int lane = threadIdx.x;
    if (node >= N_NODES) return;
    float4 h  = ((const float4*)(H + (size_t)node * DIM))[lane];
    float4 av = ((const float4*)a)[lane];
    float  s  = h.x * av.x + h.y * av.y + h.z * av.z + h.w * av.w;
#pragma unroll
    for (int m = 16; m; m >>= 1) s += __shfl_xor(s, m, 32);
    if (lane == 0) out[node] = s;
}

// ================= per-layer scratch init ==================
__global__ void init_kernel(float* __restrict__ agg, float* __restrict__ denom,
                            unsigned* __restrict__ mEnc)
{
    const int t = blockIdx.x * 256 + threadIdx.x;
    if (t < N_NODES * DIM) agg[t] = 0.0f;
    if (t < N_NODES) { denom[t] = 0.0f; mEnc[t] = ENC_NEG_INF; }
}

// ================= edge pass 1: logits + segment max ==================
__global__ void edge_logits_kernel(
    const int* __restrict__ src, const int* __restrict__ dst,
    const float* __restrict__ als, const float* __restrict__ ald,
    float* __restrict__ logits, unsigned* __restrict__ mEnc)
{
    const int e = blockIdx.x * 256 + threadIdx.x;
    if (e >= N_EDGES) return;
    float l = als[src[e]] + ald[dst[e]];
    l = l > 0.0f ? l : 0.2f * l;          // leaky_relu, slope 0.2
    logits[e] = l;
    atomicMax(&mEnc[dst[e]], encf(l));
}

// ================= edge pass 2: exp + segment sum ==================
__global__ void edge_exp_kernel(
    const int* __restrict__ dst, const float* __restrict__ logits,
    const unsigned* __restrict__ mEnc, float* __restrict__ ebuf,
    float* __restrict__ denom)
{
    const int e = blockIdx.x * 256 + threadIdx.x;
    if (e >= N_EDGES) return;
    const int d = dst[e];
    float ev = expf(logits[e] - decf(mEnc[d]));
    ebuf[e] = ev;
    atomicAdd(&denom[d], ev);
}

// ================= edge pass 3: weighted scatter (wave per edge) ==================
__global__ void edge_scatter_kernel(
    const int* __restrict__ src, const int* __restrict__ dst,
    const float* __restrict__ ebuf, const float* __restrict__ denom,
    const float* __restrict__ hsrc, float* __restrict__ agg)
{
    const long long t = (long long)blockIdx.x * 256 + threadIdx.x;
    const int e = (int)(t >> 5);
    if (e >= N_EDGES) return;
    const int lane = (int)(t & 31);
    const int s = src[e], d = dst[e];
    const float alpha = ebuf[e] / denom[d];
    float4 hv = ((const float4*)(hsrc + (size_t)s * DIM))[lane];
    float* o = agg + (size_t)d * DIM + lane * 4;
    atomicAdd(o + 0, hv.x * alpha);
    atomicAdd(o + 1, hv.y * alpha);
    atomicAdd(o + 2, hv.z * alpha);
    atomicAdd(o + 3, hv.w * alpha);
}

// ================= combine: out = [relu](agg + bc + lin) ==================
__global__ void combine_kernel(const float* __restrict__ agg, const float* __restrict__ bc,
                               const float* __restrict__ lin, float* __restrict__ out, int relu)
{
    const int t = blockIdx.x * 256 + threadIdx.x;
    if (t >= N_NODES * DIM) return;
    float v = agg[t] + bc[t & (DIM - 1)] + lin[t];
    if (relu) v = fmaxf(v, 0.0f);
    out[t] = v;
}

// =====================================================================
extern "C" void kernel_launch(void* const* d_in, const int* in_sizes, int n_in,
                              void* d_out, int out_size, void* d_ws, size_t ws_size,
                              hipStream_t stream)
{
    const float* x  = (const float*)d_in[0];
    const int*   ei = (const int*)d_in[1];
    const int* src = ei;
    const int* dst = ei + N_EDGES;

    const float *w0_src = (const float*)d_in[2],  *w0_dst = (const float*)d_in[3];
    const float *a0_src = (const float*)d_in[4],  *a0_dst = (const float*)d_in[5];
    const float *bc0    = (const float*)d_in[6];
    const float *wl0    = (const float*)d_in[7],  *bl0    = (const float*)d_in[8];
    const float *w1     = (const float*)d_in[9];
    const float *a1_src = (const float*)d_in[10], *a1_dst = (const float*)d_in[11];
    const float *bc1    = (const float*)d_in[12];
    const float *wl1    = (const float*)d_in[13], *bl1    = (const float*)d_in[14];
    const float *w2     = (const float*)d_in[15];
    const float *a2_src = (const float*)d_in[16], *a2_dst = (const float*)d_in[17];
    const float *bc2    = (const float*)d_in[18];
    const float *wl2    = (const float*)d_in[19], *bl2    = (const float*)d_in[20];
    const float *wh     = (const float*)d_in[21], *bh     = (const float*)d_in[22];

    // ---- workspace layout ----
    float* ws = (float*)d_ws;
    float*    h      = ws;                         // N*128
    float*    hsrc   = h    + (size_t)N_NODES * DIM;
    float*    hdst   = hsrc + (size_t)N_NODES * DIM;
    float*    lin    = hdst + (size_t)N_NODES * DIM;
    float*    agg    = lin  + (size_t)N_NODES * DIM;
    float*    als    = agg  + (size_t)N_NODES * DIM;   // N
    float*    ald    = als  + N_NODES;
    float*    denom  = ald  + N_NODES;
    unsigned* mEnc   = (unsigned*)(denom + N_NODES);
    float*    logits = (float*)(mEnc + N_NODES);       // E
    float*    ebuf   = logits + N_EDGES;               // E

    const dim3 gemmBlk(32, DIM / 16);
    const dim3 headBlk(32, HEADD / 16);
    const dim3 dotBlk(32, 8);
    const int  gemmGrid = N_NODES / 16;                 // 3125
    const int  dotGrid  = (N_NODES + 7) / 8;
    const int  ndGrid   = (N_NODES * DIM + 255) / 256;  // 25000
    const int  eGrid    = (N_EDGES + 255) / 256;
    const int  sGrid    = (int)(((long long)N_EDGES * 32 + 255) / 256);

    // -------- layer 0 --------
    gemm_wmma<DIM><<<gemmGrid, gemmBlk, 0, stream>>>(x, w0_src, nullptr, hsrc);
    gemm_wmma<DIM><<<gemmGrid, gemmBlk, 0, stream>>>(x, w0_dst, nullptr, hdst);
    gemm_wmma<DIM><<<gemmGrid, gemmBlk, 0, stream>>>(x, wl0, bl0, lin);
    dot_kernel<<<dotGrid, dotBlk, 0, stream>>>(hsrc, a0_src, als);
    dot_kernel<<<dotGrid, dotBlk, 0, stream>>>(hdst, a0_dst, ald);
    init_kernel<<<ndGrid, 256, 0, stream>>>(agg, denom, mEnc);
    edge_logits_kernel<<<eGrid, 256, 0, stream>>>(src, dst, als, ald, logits, mEnc);
    edge_exp_kernel<<<eGrid, 256, 0, stream>>>(dst, logits, mEnc, ebuf, denom);
    edge_scatter_kernel<<<sGrid, 256, 0, stream>>>(src, dst, ebuf, denom, hsrc, agg);
    combine_kernel<<<ndGrid, 256, 0, stream>>>(agg, bc0, lin, h, 1);

    // -------- layer 1 (shared src/dst transform) --------
    gemm_wmma<DIM><<<gemmGrid, gemmBlk, 0, stream>>>(h, w1, nullptr, hsrc);
    gemm_wmma<DIM><<<gemmGrid, gemmBlk, 0, stream>>>(h, wl1, bl1, lin);
    dot_kernel<<<dotGrid, dotBlk, 0, stream>>>(hsrc, a1_src, als);
    dot_kernel<<<dotGrid, dotBlk, 0, stream>>>(hsrc, a1_dst, ald);
    init_kernel<<<ndGrid, 256, 0, stream>>>(agg, denom, mEnc);
    edge_logits_kernel<<<eGrid, 256, 0, stream>>>(src, dst, als, ald, logits, mEnc);
    edge_exp_kernel<<<eGrid, 256, 0, stream>>>(dst, logits, mEnc, ebuf, denom);
    edge_scatter_kernel<<<sGrid, 256, 0, stream>>>(src, dst, ebuf, denom, hsrc, agg);
    combine_kernel<<<ndGrid, 256, 0, stream>>>(agg, bc1, lin, h, 1);

    // -------- layer 2 (no relu) --------
    gemm_wmma<DIM><<<gemmGrid, gemmBlk, 0, stream>>>(h, w2, nullptr, hsrc);
    gemm_wmma<DIM><<<gemmGrid, gemmBlk, 0, stream>>>(h, wl2, bl2, lin);
    dot_kernel<<<dotGrid, dotBlk, 0, stream>>>(hsrc, a2_src, als);
    dot_kernel<<<dotGrid, dotBlk, 0, stream>>>(hsrc, a2_dst, ald);
    init_kernel<<<ndGrid, 256, 0, stream>>>(agg, denom, mEnc);
    edge_logits_kernel<<<eGrid, 256, 0, stream>>>(src, dst, als, ald, logits, mEnc);
    edge_exp_kernel<<<eGrid, 256, 0, stream>>>(dst, logits, mEnc, ebuf, denom);
    edge_scatter_kernel<<<sGrid, 256, 0, stream>>>(src, dst, ebuf, denom, hsrc, agg);
    combine_kernel<<<ndGrid, 256, 0, stream>>>(agg, bc2, lin, h, 0);

    // -------- head --------
    gemm_wmma<HEADD><<<gemmGrid, headBlk, 0, stream>>>(h, wh, bh, (float*)d_out);
}